// YOLOLayer_34565896798706
// MI455X (gfx1250) — compile-verified
//
#include <hip/hip_runtime.h>
#include <cstdint>

// YOLO decode layer for MI455X (gfx1250).
// Memory-bound: ~188.5 MB in + ~188.5 MB out => ~16 us at 23.3 TB/s roofline.
// Strategy: stage the (channel-major -> channel-minor) transpose through LDS
// using CDNA5 async global->LDS b64 copies (ASYNCcnt-tracked), compute the
// anchor-axis softmax once per (class,pos) in LDS, then stream coalesced
// nontemporal stores of contiguous 85-float output rows.

#define BB   32
#define AA   3
#define CC   80
#define GG   76
#define GG2  (GG * GG)          // 5776
#define CHN  (AA * (CC + 5))    // 255
#define TP   38                 // spatial positions per tile (half grid row)
#define TPQ  (TP / 2)           // 19 b64 transfers per channel row
#define TILE_F (CHN * TP)       // 9690 floats
#define TILE_Q (CHN * TPQ)      // 4845 b64 transfers
#define DEN_F  (CC * TP)        // 3040 floats
#define SMEM_F (TILE_F + DEN_F) // 12730 floats = 50920 B LDS
#define NBLK (BB * GG * 2)      // 4864 blocks
#define NTHR 256                // 8 wave32s

__device__ __forceinline__ float fast_sigmoid(float x) {
    // v_exp_f32 + v_rcp_f32
    return __builtin_amdgcn_rcpf(1.0f + __expf(-x));
}

__global__ void __launch_bounds__(NTHR)
yolo_decode(const float* __restrict__ in, float* __restrict__ out) {
    extern __shared__ float smem[];
    float* tile = smem;            // [CHN][TP] raw; class chans overwritten w/ exp
    float* dinv = smem + TILE_F;   // [CC][TP]  1 / sum_a exp

    const int bi   = blockIdx.x;
    const int b    = bi / (GG * 2);
    const int r    = bi - b * (GG * 2);
    const int i    = r >> 1;          // grid row
    const int h    = r & 1;           // half-row index
    const int pos0 = i * GG + h * TP; // spatial base within the grid

    const float* src = in + (size_t)b * CHN * GG2 + pos0;

    // ---- Phase 1: async copy (CHN x TP) tile, global -> LDS, b64 wide ----
    // Per-lane global_load_async_to_lds_b64 (GVS mode): VDST = LDS byte
    // offset, VADDR = per-lane byte offset, SADDR = uniform 64-bit base.
    // All addresses are 8-byte aligned: global rows start at multiples of
    // 152 B from an aligned base; LDS rows are 152 B apart.
    for (int e = threadIdx.x; e < TILE_Q; e += NTHR) {
        const int ch = e / TPQ;
        const int k  = e - ch * TPQ;                          // b64 slot in row
        const uint32_t voff = (uint32_t)(ch * GG2 * 4 + k * 8); // bytes from src
        const uint32_t loff = (uint32_t)((ch * TP + k * 2) * 4); // LDS byte off
        asm volatile("global_load_async_to_lds_b64 %0, %1, %2"
                     :
                     : "v"(loff), "v"(voff), "s"(src)
                     : "memory");
    }
    asm volatile("s_wait_asynccnt 0" ::: "memory"); // drain this wave's ASYNCcnt
    __syncthreads();                                // make LDS visible block-wide

    // ---- Phase 2: anchor-axis softmax pieces -----------------------------
    // For each (class c, pos j): exponentiate the 3 anchor logits (max-sub
    // for stability) in place, and store reciprocal of their sum.
    for (int e = threadIdx.x; e < DEN_F; e += NTHR) {
        const int c = e / TP;
        const int j = e - c * TP;
        const float v0 = tile[(5 + c) * TP + j];    // a=0: ch = 0*85+5+c
        const float v1 = tile[(90 + c) * TP + j];   // a=1: ch = 1*85+5+c
        const float v2 = tile[(175 + c) * TP + j];  // a=2: ch = 2*85+5+c
        const float m  = fmaxf(v0, fmaxf(v1, v2));
        const float e0 = __expf(v0 - m);
        const float e1 = __expf(v1 - m);
        const float e2 = __expf(v2 - m);
        tile[(5 + c) * TP + j]   = e0;
        tile[(90 + c) * TP + j]  = e1;
        tile[(175 + c) * TP + j] = e2;
        dinv[e] = __builtin_amdgcn_rcpf(e0 + e1 + e2);
    }
    __syncthreads();

    // ---- Phase 3: decode + coalesced nontemporal stores ------------------
    // Output element (b, a, pos0+j, c) at ((b*3+a)*5776 + pos0 + j)*85 + c.
    // Consecutive e -> consecutive c => contiguous 85-float runs per box.
    const int OUT_F = AA * TP * 85;   // 9690
    for (int e = threadIdx.x; e < OUT_F; e += NTHR) {
        const int c = e % 85;
        const int t = e / 85;
        const int j = t % TP;
        const int a = t / TP;
        const float v = tile[(a * 85 + c) * TP + j];
        float res;
        if (c >= 5) {
            res = v * dinv[(c - 5) * TP + j];            // softmax over anchors
        } else if (c == 0) {
            res = (fast_sigmoid(v) + (float)(h * TP + j)) * 8.0f; // bx * stride
        } else if (c == 1) {
            res = (fast_sigmoid(v) + (float)i) * 8.0f;            // by * stride
        } else if (c == 2) {
            const float aw = (a == 0) ? 10.0f : ((a == 1) ? 16.0f : 33.0f);
            res = __expf(v) * aw;                        // (anchor/8)*exp(w)*8
        } else if (c == 3) {
            const float ah = (a == 0) ? 13.0f : ((a == 1) ? 30.0f : 23.0f);
            res = __expf(v) * ah;
        } else { // c == 4
            res = fast_sigmoid(v);                       // objectness
        }
        const size_t oidx = ((size_t)((b * AA + a) * GG2 + pos0 + j)) * 85 + c;
        __builtin_nontemporal_store(res, &out[oidx]);
    }
}

extern "C" void kernel_launch(void* const* d_in, const int* in_sizes, int n_in,
                              void* d_out, int out_size, void* d_ws, size_t ws_size,
                              hipStream_t stream) {
    const float* pred = (const float*)d_in[0];
    float* out = (float*)d_out;
    yolo_decode<<<NBLK, NTHR, SMEM_F * sizeof(float), stream>>>(pred, out);
}